// TopkWeightClusterLoss_82059645157779
// MI455X (gfx1250) — compile-verified
//
#include <hip/hip_runtime.h>
#include <cstdint>

// CDNA5 async global->LDS copy (ASYNCcnt-tracked). vdst = LDS byte address,
// vaddr = 64-bit global address, saddr = off. Each lane moves 16 bytes.
#define ASYNC_LOAD_B128(ldsaddr, gaddr)                                        \
  asm volatile("global_load_async_to_lds_b128 %0, %1, off"                     \
               :: "v"(ldsaddr), "v"(gaddr) : "memory")

#define WAIT_ASYNCCNT(n)                                                       \
  asm volatile("s_wait_asynccnt " #n ::: "memory")

static constexpr int NCOL = 4096;
static constexpr int NROW = 4096;
static constexpr int THREADS = 256;          // 8 wave32 per block
static constexpr int ROWS_PER_BLOCK = 4;
static constexpr int NBLOCKS = NROW / ROWS_PER_BLOCK;   // 1024

// Per-element loss term: sum(top4) - dot(top4, softmax(top4)).
// Top-4 of the 15 valley-shaped distances via the bitonic half-cleaner
// identity: top4_i = max(d_i, d_{11+i}) over centers {-7s..-4s} x {4s..7s}.
__device__ __forceinline__ float elem_loss(float w, float c4, float c5,
                                           float c6, float c7) {
  float a0 = w + c7, a1 = w + c6, a2 = w + c5, a3 = w + c4;   // left arm
  float b0 = w - c4, b1 = w - c5, b2 = w - c6, b3 = w - c7;   // right arm
  a0 *= a0; a1 *= a1; a2 *= a2; a3 *= a3;
  b0 *= b0; b1 *= b1; b2 *= b2; b3 *= b3;
  float t0 = fmaxf(a0, b0);
  float t1 = fmaxf(a1, b1);
  float t2 = fmaxf(a2, b2);
  float t3 = fmaxf(a3, b3);
  float m = fmaxf(t0, t3);                    // valley max is at an end
  float x0 = __expf(t0 - m);
  float x1 = __expf(t1 - m);
  float x2 = __expf(t2 - m);
  float x3 = __expf(t3 - m);
  float se = (x0 + x1) + (x2 + x3);
  float tt = fmaf(t0, x0, fmaf(t1, x1, fmaf(t2, x2, t3 * x3)));
  float sv = (t0 + t1) + (t2 + t3);
  return sv - __fdividef(tt, se);
}

__global__ void __launch_bounds__(THREADS)
topk_loss_kernel(const float* __restrict__ W, const float* __restrict__ S,
                 float* __restrict__ partials) {
  __shared__ __align__(16) float buf[2][NCOL];   // double-buffered row stage
  __shared__ float red[THREADS];

  const int t = threadIdx.x;
  const int wave = t >> 5;
  const int lane = t & 31;
  // Wave w owns columns [w*512, w*512+512); lane stages+consumes 4 float4s.
  const int colBase = wave * 512 + lane * 4;     // float index
  const size_t row0 = (size_t)blockIdx.x * ROWS_PER_BLOCK;

  const uint32_t ldsBase0 = (uint32_t)(size_t)&buf[0][colBase];
  const uint32_t ldsBase1 = (uint32_t)(size_t)&buf[1][colBase];

  // Prologue: stage row 0 into buffer 0 (4 async b128 per wave = 2KB/wave).
  {
    uint64_t ga = (uint64_t)(size_t)(W + row0 * NCOL + colBase);
    ASYNC_LOAD_B128(ldsBase0,         ga);
    ASYNC_LOAD_B128(ldsBase0 +  512u, ga +  512u);
    ASYNC_LOAD_B128(ldsBase0 + 1024u, ga + 1024u);
    ASYNC_LOAD_B128(ldsBase0 + 1536u, ga + 1536u);
  }

  float acc = 0.0f;
#pragma unroll
  for (int r = 0; r < ROWS_PER_BLOCK; ++r) {
    if (r + 1 < ROWS_PER_BLOCK) {
      // Prefetch next row into the other buffer, then wait until only those
      // 4 newest async ops remain outstanding (async loads retire in order),
      // guaranteeing the current buffer is fully in LDS.
      const uint32_t lnext = (r & 1) ? ldsBase0 : ldsBase1;
      uint64_t ga = (uint64_t)(size_t)(W + (row0 + r + 1) * NCOL + colBase);
      ASYNC_LOAD_B128(lnext,         ga);
      ASYNC_LOAD_B128(lnext +  512u, ga +  512u);
      ASYNC_LOAD_B128(lnext + 1024u, ga + 1024u);
      ASYNC_LOAD_B128(lnext + 1536u, ga + 1536u);
      WAIT_ASYNCCNT(4);
    } else {
      WAIT_ASYNCCNT(0);
    }

    const float sc = S[row0 + r];                 // block-uniform scalar load
    const float c4 = 4.0f * sc, c5 = 5.0f * sc;
    const float c6 = 6.0f * sc, c7 = 7.0f * sc;

    const float* bp = (r & 1) ? buf[1] : buf[0];
#pragma unroll
    for (int q = 0; q < 4; ++q) {
      const float4 v = *(const float4*)(bp + colBase + q * 128);
      acc += elem_loss(v.x, c4, c5, c6, c7);
      acc += elem_loss(v.y, c4, c5, c6, c7);
      acc += elem_loss(v.z, c4, c5, c6, c7);
      acc += elem_loss(v.w, c4, c5, c6, c7);
    }
  }

  // Deterministic in-block tree reduction.
  red[t] = acc;
  __syncthreads();
#pragma unroll
  for (int off = THREADS / 2; off > 0; off >>= 1) {
    if (t < off) red[t] += red[t + off];
    __syncthreads();
  }
  if (t == 0) partials[blockIdx.x] = red[0];
}

__global__ void __launch_bounds__(THREADS)
final_reduce_kernel(const float* __restrict__ partials, float* __restrict__ out,
                    int n) {
  __shared__ float red[THREADS];
  float a = 0.0f;
  for (int i = threadIdx.x; i < n; i += THREADS) a += partials[i];
  red[threadIdx.x] = a;
  __syncthreads();
#pragma unroll
  for (int off = THREADS / 2; off > 0; off >>= 1) {
    if (threadIdx.x < off) red[threadIdx.x] += red[threadIdx.x + off];
    __syncthreads();
  }
  if (threadIdx.x == 0) out[0] = red[0] * (0.01f / 3.0f);   // COEFF/(TOPK-1)
}

extern "C" void kernel_launch(void* const* d_in, const int* in_sizes, int n_in,
                              void* d_out, int out_size, void* d_ws,
                              size_t ws_size, hipStream_t stream) {
  (void)in_sizes; (void)n_in; (void)out_size; (void)ws_size;
  const float* W = (const float*)d_in[0];   // weight [4096,4096] f32
  const float* S = (const float*)d_in[1];   // scale  [4096,1]   f32
  float* partials = (float*)d_ws;           // NBLOCKS floats (4 KB)

  topk_loss_kernel<<<NBLOCKS, THREADS, 0, stream>>>(W, S, partials);
  final_reduce_kernel<<<1, THREADS, 0, stream>>>(partials, (float*)d_out,
                                                 NBLOCKS);
}